// GPRGNNConv_68143951118623
// MI455X (gfx1250) — compile-verified
//
#include <hip/hip_runtime.h>
#include <hip/hip_bf16.h>
#include <math.h>

typedef float v2f __attribute__((ext_vector_type(2)));
typedef float v8f __attribute__((ext_vector_type(8)));

#define FEAT 64
#define LDS_STRIDE 66   // pad to avoid 16-way LDS bank conflicts on column reads

// ---------------------------------------------------------------------------
// Kernel 0: softmax over gamma (K+1 = 11 values) -> g[] in workspace
// ---------------------------------------------------------------------------
__global__ void softmax_small_kernel(const float* __restrict__ gamma,
                                     float* __restrict__ g, int n) {
    if (blockIdx.x == 0 && threadIdx.x == 0) {
        float m = -1e30f;
        for (int i = 0; i < n; ++i) m = fmaxf(m, gamma[i]);
        float s = 0.f;
        for (int i = 0; i < n; ++i) { float e = __expf(gamma[i] - m); g[i] = e; s += e; }
        float inv = 1.f / s;
        for (int i = 0; i < n; ++i) g[i] *= inv;
    }
}

// ---------------------------------------------------------------------------
// Kernel 1: y = x @ W^T + b  via V_WMMA_F32_16X16X4_F32
// One wave computes a 16x64 output tile: 4 accumulators (16x16), K-loop 64/4.
// ---------------------------------------------------------------------------
__global__ __launch_bounds__(256) void gemm_wmma_kernel(
    const float* __restrict__ x, const float* __restrict__ W,
    const float* __restrict__ bias, float* __restrict__ y, int nrows) {

    __shared__ float Wl[FEAT * LDS_STRIDE];
    __shared__ float bl[FEAT];

    const int t = threadIdx.x;
    // Stage W (row-major, padded stride) and bias into LDS.
    for (int i = t; i < FEAT * FEAT; i += 256) {
        int r = i >> 6, c = i & 63;
        Wl[r * LDS_STRIDE + c] = W[i];
    }
    if (t < FEAT) bl[t] = bias[t];
    __syncthreads();

    const int wave = t >> 5;
    const int lane = t & 31;
    const int nTiles = (nrows + 15) >> 4;
    const int tile = blockIdx.x * 8 + wave;
    if (tile >= nTiles) return;
    const int rowBase = tile << 4;

    // A-frag addressing: lanes 0-15 -> M=lane, K={k0,k0+1}; lanes 16-31 -> K={k0+2,k0+3}
    const int mA = lane & 15;
    const int kOff = (lane >> 4) << 1;          // 0 or 2
    int rA = rowBase + mA;
    if (rA >= nrows) rA = nrows - 1;            // clamp (N=100000 is exactly divisible)
    const float* __restrict__ xrow = x + (size_t)rA * FEAT;

    // B-frag addressing: B[k][j] = W[j][k]; lanes 0-15 -> j = n0+lane, K={k0,k0+1}
    const int jb = lane & 15;

    v8f acc[4];
    #pragma unroll
    for (int n = 0; n < 4; ++n) acc[n] = (v8f){0.f,0.f,0.f,0.f,0.f,0.f,0.f,0.f};

    #pragma unroll
    for (int k0 = 0; k0 < FEAT; k0 += 4) {
        v2f a = *(const v2f*)(xrow + k0 + kOff);
        #pragma unroll
        for (int n = 0; n < 4; ++n) {
            const float* bp = &Wl[(n * 16 + jb) * LDS_STRIDE + k0 + kOff];
            v2f bfrag = { bp[0], bp[1] };
            acc[n] = __builtin_amdgcn_wmma_f32_16x16x4_f32(
                /*neg_a=*/false, a, /*neg_b=*/false, bfrag,
                /*c_mod=*/(short)0, acc[n], /*reuse_a=*/false, /*reuse_b=*/false);
        }
    }

    // C/D layout: lane&15 = N, VGPR v holds M=v (lanes 0-15) / M=v+8 (lanes 16-31)
    const int mOff = (lane >> 4) << 3;          // 0 or 8
    #pragma unroll
    for (int n = 0; n < 4; ++n) {
        const int col = n * 16 + jb;
        const float bb = bl[col];
        #pragma unroll
        for (int v = 0; v < 8; ++v) {
            const int row = rowBase + mOff + v;
            if (row < nrows) y[(size_t)row * FEAT + col] = acc[n][v] + bb;
        }
    }
}

// ---------------------------------------------------------------------------
// Kernel 2: h = g[0]*cur ; zbuf = 0   (vectorized float4 over N*64 elements)
// ---------------------------------------------------------------------------
__global__ __launch_bounds__(256) void init_h_kernel(
    const float* __restrict__ cur, float* __restrict__ h,
    float* __restrict__ zbuf, const float* __restrict__ g, unsigned n4) {
    unsigned i = blockIdx.x * 256u + threadIdx.x;
    if (i >= n4) return;
    const float g0 = g[0];
    float4 v = ((const float4*)cur)[i];
    ((float4*)h)[i] = make_float4(g0 * v.x, g0 * v.y, g0 * v.z, g0 * v.w);
    ((float4*)zbuf)[i] = make_float4(0.f, 0.f, 0.f, 0.f);
}

// ---------------------------------------------------------------------------
// Kernel 3: SpMM scatter: acc[row[e]][:] += w[e] * cur[col[e]][:]
// 16 threads per edge, float4 gather + 4 global f32 atomics each.
// ---------------------------------------------------------------------------
__global__ __launch_bounds__(256) void scatter_kernel(
    const int* __restrict__ rowI, const int* __restrict__ colI,
    const float* __restrict__ ew, const float* __restrict__ cur,
    float* __restrict__ acc, int E) {
    unsigned tid = blockIdx.x * 256u + threadIdx.x;
    int e = (int)(tid >> 4);
    if (e >= E) return;
    const int f = (int)(tid & 15u) << 2;
    const int r = rowI[e];
    const int c = colI[e];
    const float w = ew[e];
    const float4 v = *(const float4*)(cur + (size_t)c * FEAT + f);
    float* dst = acc + (size_t)r * FEAT + f;
    atomicAdd(dst + 0, w * v.x);
    atomicAdd(dst + 1, w * v.y);
    atomicAdd(dst + 2, w * v.z);
    atomicAdd(dst + 3, w * v.w);
}

// ---------------------------------------------------------------------------
// Kernel 4: h += g[k]*newbuf ; optionally zero the retiring buffer
// ---------------------------------------------------------------------------
__global__ __launch_bounds__(256) void axpy_zero_kernel(
    float* __restrict__ h, const float* __restrict__ newbuf,
    float* __restrict__ zbuf, const float* __restrict__ g, int kidx,
    unsigned n4, int doZero) {
    unsigned i = blockIdx.x * 256u + threadIdx.x;
    if (i >= n4) return;
    const float gk = g[kidx];
    float4 v = ((const float4*)newbuf)[i];
    float4 hv = ((float4*)h)[i];
    ((float4*)h)[i] = make_float4(hv.x + gk * v.x, hv.y + gk * v.y,
                                  hv.z + gk * v.z, hv.w + gk * v.w);
    if (doZero) ((float4*)zbuf)[i] = make_float4(0.f, 0.f, 0.f, 0.f);
}

// ---------------------------------------------------------------------------
extern "C" void kernel_launch(void* const* d_in, const int* in_sizes, int n_in,
                              void* d_out, int out_size, void* d_ws, size_t ws_size,
                              hipStream_t stream) {
    const float* x     = (const float*)d_in[0];   // [N, 64]
    const float* ew    = (const float*)d_in[1];   // [E]
    const float* W     = (const float*)d_in[2];   // [64, 64]
    const float* bias  = (const float*)d_in[3];   // [64]
    const float* gamma = (const float*)d_in[4];   // [K+1]
    const int*   eidx  = (const int*)d_in[5];     // [2, E]

    const int N   = in_sizes[0] / FEAT;
    const int E   = in_sizes[1];
    const int Kp1 = in_sizes[4];

    float* g    = (float*)d_ws;                               // 11 floats (pad 256B)
    float* bufA = (float*)((char*)d_ws + 256);                // [N,64]
    float* bufB = bufA + (size_t)N * FEAT;                    // [N,64]
    float* h    = (float*)d_out;

    const unsigned n4 = (unsigned)((size_t)N * FEAT / 4);     // float4 count
    const unsigned ewBlocks = (n4 + 255u) / 256u;

    // 1) softmax(gamma)
    softmax_small_kernel<<<1, 32, 0, stream>>>(gamma, g, Kp1);

    // 2) bufA = x @ W^T + b  (WMMA)
    {
        int nTiles = (N + 15) >> 4;
        int blocks = (nTiles + 7) >> 3;       // 8 waves (tiles) per block
        gemm_wmma_kernel<<<blocks, 256, 0, stream>>>(x, W, bias, bufA, N);
    }

    // 3) h = g0 * bufA ; bufB = 0
    init_h_kernel<<<ewBlocks, 256, 0, stream>>>(bufA, h, bufB, g, n4);

    // 4) K propagation steps with ping-pong buffers
    const int* rowI = eidx;
    const int* colI = eidx + E;
    float* cur = bufA;
    float* nxt = bufB;
    const unsigned scatterThreads = (unsigned)E * 16u;
    const unsigned scatterBlocks = (scatterThreads + 255u) / 256u;

    for (int k = 1; k < Kp1; ++k) {
        scatter_kernel<<<scatterBlocks, 256, 0, stream>>>(rowI, colI, ew, cur, nxt, E);
        axpy_zero_kernel<<<ewBlocks, 256, 0, stream>>>(h, nxt, cur, g, k, n4,
                                                       (k + 1 < Kp1) ? 1 : 0);
        float* tmp = cur; cur = nxt; nxt = tmp;
    }
}